// dummyCEVAE_31430570672736
// MI455X (gfx1250) — compile-verified
//
#include <hip/hip_runtime.h>
#include <hip/hip_bf16.h>

typedef __attribute__((ext_vector_type(2))) float v2f;
typedef __attribute__((ext_vector_type(8))) float v8f;

#define CEVAE_D 20

// ---------------------------------------------------------------------------
// z_logits via WMMA index dot-product + scalar heads + pz passthrough.
// One wave handles 16 rows:
//   A (16x4 f32): lane L -> row M=L&15, K-half = L>>4  (VGPR0=K0/K2, VGPR1=K1/K3)
//   B (4x16 f32): B[k][*] = 2^(19-k)  (replicated across all 16 columns)
//   C (16x16 f32): every column of row m == combo index of row m (exact in f32,
//   all partial sums are integers < 2^24)
// ---------------------------------------------------------------------------
__global__ __launch_bounds__(256)
void cevae_z_heads_kernel(const float* __restrict__ x,
                          const float* __restrict__ t,
                          const float* __restrict__ y,
                          const float* __restrict__ qz_w,
                          const float* __restrict__ qz_b,
                          const float* __restrict__ t_w_p,
                          const float* __restrict__ t_b_p,
                          const float* __restrict__ y0_w_p,
                          const float* __restrict__ y0_b_p,
                          const float* __restrict__ y1_w_p,
                          const float* __restrict__ y1_b_p,
                          const float* __restrict__ pz_p,
                          float* __restrict__ out_z,
                          float* __restrict__ out_t0,
                          float* __restrict__ out_t1,
                          float* __restrict__ out_y0,
                          float* __restrict__ out_y1,
                          float* __restrict__ out_pz,
                          int N, int D)
{
    if (blockIdx.x == 0 && threadIdx.x == 0) out_pz[0] = pz_p[0];

    const int lane     = threadIdx.x & 31;
    const int wave     = blockIdx.x * (blockDim.x >> 5) + (threadIdx.x >> 5);
    const int row_base = wave * 16;
    if (row_base >= N) return;                       // wave-uniform exit

    const bool full_tile = (row_base + 16 <= N) && (D == CEVAE_D);

    float idx_f = 0.0f;
    if (full_tile) {
        // ---- WMMA index computation, EXEC is all-ones here (uniform branch)
        const int m  = lane & 15;
        const int kh = lane >> 4;                    // K-half: 0 -> K{0,1}, 1 -> K{2,3}
        const float* xrow = x + (size_t)(row_base + m) * CEVAE_D + 2 * kh;

        v8f c = {};                                  // accumulator starts at 0
#pragma unroll
        for (int k0 = 0; k0 < CEVAE_D; k0 += 4) {
            v2f a = *(const v2f*)(xrow + k0);        // A frag: x[m][k0+2kh .. +1]
            v2f b;
            const int kk = CEVAE_D - 1 - (k0 + 2 * kh);
            b.x = (float)(1u << kk);                 // 2^(19-k)
            b.y = (float)(1u << (kk - 1));
            c = __builtin_amdgcn_wmma_f32_16x16x4_f32(
                    false, a, false, b, (short)0, c, false, false);
        }
        // C layout: lanes 0-15 vgpr i -> row i ; lanes 16-31 vgpr i -> row 8+i.
        // Branchless binary select tree (7 v_cndmask, no exec-mask branches).
        const int s = lane & 7;
        const bool b4 = (s & 4) != 0;
        const bool b2 = (s & 2) != 0;
        const bool b1 = (s & 1) != 0;
        float a0 = b4 ? c[4] : c[0];
        float a1 = b4 ? c[5] : c[1];
        float a2 = b4 ? c[6] : c[2];
        float a3 = b4 ? c[7] : c[3];
        float m0 = b2 ? a2 : a0;
        float m1 = b2 ? a3 : a1;
        idx_f    = b1 ? m1 : m0;
    }

    // One gather lane per row: lanes 0-7 -> rows +0..7, lanes 16-23 -> rows +8..15
    if ((lane & 15) < 8) {
        const int r = row_base + ((lane >> 4) << 3) + (lane & 7);
        if (r < N) {
            int idx;
            if (full_tile) {
                idx = (int)idx_f;                    // exact integer < 2^20
            } else {                                 // ragged-tail / generic-D path
                const float* xr = x + (size_t)r * D;
                idx = 0;
                for (int j = 0; j < D; ++j)
                    idx = (idx << 1) | (xr[j] != 0.0f ? 1 : 0);
            }
            const float tv = t[r];
            idx += ((int)tv) << D;                   // t is the MSB of the combo

            const float zv = qz_w[idx] * y[r] + qz_b[idx];
            out_z[r] = zv;

            const float tb  = t_b_p[0],  tw  = t_w_p[0];
            const float y0w = y0_w_p[0], y0b = y0_b_p[0];
            const float y1w = y1_w_p[0], y1b = y1_b_p[0];
            out_t0[r] = tb;                          // z=0 head
            out_t1[r] = tw + tb;                     // z=1 head
            out_y0[r] = y1b * tv + y0b * (1.0f - tv);
            out_y1[r] = (y1w + y1b) * tv + (y0w + y0b) * (1.0f - tv);
        }
    }
}

// ---------------------------------------------------------------------------
// Broadcast fill of x_logits (row-independent pattern of length D).
// D % 4 == 0 -> pattern period is D/4 float4s; one b128 store per thread.
// ---------------------------------------------------------------------------
__global__ __launch_bounds__(256)
void cevae_fill_vec4_kernel(float* __restrict__ out,
                            const float* __restrict__ dx_w,
                            const float* __restrict__ dx_b,
                            float wscale, unsigned int nvec4, int D)
{
    __shared__ __align__(16) float pat[32];
    if ((int)threadIdx.x < D)
        pat[threadIdx.x] = dx_b[threadIdx.x] + wscale * dx_w[threadIdx.x];
    __syncthreads();

    const unsigned int i = blockIdx.x * blockDim.x + threadIdx.x;
    if (i >= nvec4) return;
    const unsigned int period = (unsigned int)(D >> 2);      // 5 for D=20
    const float4* p4 = (const float4*)pat;
    ((float4*)out)[i] = p4[i % period];
}

// Generic scalar fallback (used if D % 4 != 0 or D > 32).
__global__ __launch_bounds__(256)
void cevae_fill_scalar_kernel(float* __restrict__ out,
                              const float* __restrict__ dx_w,
                              const float* __restrict__ dx_b,
                              float wscale, unsigned long long total, int D)
{
    unsigned long long i = blockIdx.x * (unsigned long long)blockDim.x + threadIdx.x;
    if (i < total) {
        int j = (int)(i % (unsigned long long)D);
        out[i] = dx_b[j] + wscale * dx_w[j];
    }
}

extern "C" void kernel_launch(void* const* d_in, const int* in_sizes, int n_in,
                              void* d_out, int out_size, void* d_ws, size_t ws_size,
                              hipStream_t stream)
{
    const float* x    = (const float*)d_in[0];
    const float* t    = (const float*)d_in[1];
    const float* y    = (const float*)d_in[2];
    const float* qz_w = (const float*)d_in[3];
    const float* qz_b = (const float*)d_in[4];
    const float* dx_w = (const float*)d_in[5];
    const float* dx_b = (const float*)d_in[6];
    const float* t_w  = (const float*)d_in[7];
    const float* t_b  = (const float*)d_in[8];
    const float* y0_w = (const float*)d_in[9];
    const float* y0_b = (const float*)d_in[10];
    const float* y1_w = (const float*)d_in[11];
    const float* y1_b = (const float*)d_in[12];
    const float* pz   = (const float*)d_in[13];

    const int N = in_sizes[1];          // t has N elements
    const int D = in_sizes[5];          // dx_w has D elements (20)
    const long long ND = (long long)N * (long long)D;

    float* out    = (float*)d_out;
    float* out_z  = out;
    float* out_x0 = out + N;
    float* out_x1 = out + N + ND;
    float* out_t0 = out + (long long)N + 2 * ND;
    float* out_t1 = out + 2LL * N + 2 * ND;
    float* out_y0 = out + 3LL * N + 2 * ND;
    float* out_y1 = out + 4LL * N + 2 * ND;
    float* out_pz = out + 5LL * N + 2 * ND;

    // --- z_logits (WMMA index) + t/y heads + pz: 256 thr = 8 waves = 128 rows
    {
        const int rows_per_block = 128;
        const int grid = (N + rows_per_block - 1) / rows_per_block;
        cevae_z_heads_kernel<<<grid, 256, 0, stream>>>(
            x, t, y, qz_w, qz_b, t_w, t_b, y0_w, y0_b, y1_w, y1_b, pz,
            out_z, out_t0, out_t1, out_y0, out_y1, out_pz, N, D);
    }

    // --- x_logits0 / x_logits1 broadcast fills (store-bandwidth bound)
    if ((D % 4) == 0 && D <= 32) {
        const unsigned int nvec4 = (unsigned int)(ND >> 2);
        const unsigned int grid  = (nvec4 + 255u) / 256u;
        cevae_fill_vec4_kernel<<<grid, 256, 0, stream>>>(out_x0, dx_w, dx_b, 0.0f, nvec4, D);
        cevae_fill_vec4_kernel<<<grid, 256, 0, stream>>>(out_x1, dx_w, dx_b, 1.0f, nvec4, D);
    } else {
        const unsigned long long total = (unsigned long long)ND;
        const unsigned int grid = (unsigned int)((total + 255ull) / 256ull);
        cevae_fill_scalar_kernel<<<grid, 256, 0, stream>>>(out_x0, dx_w, dx_b, 0.0f, total, D);
        cevae_fill_scalar_kernel<<<grid, 256, 0, stream>>>(out_x1, dx_w, dx_b, 1.0f, total, D);
    }
}